// SpikeActivation_61100204752915
// MI455X (gfx1250) — compile-verified
//
#include <hip/hip_runtime.h>
#include <cstdint>

// ---------------------------------------------------------------------------
// SpikeActivation scan, MI455X (gfx1250).
//
// V_t = reset(V_{t-1} + I_t),  reset(x) = |x|>1 ? 0 : x,  T=4 chunks.
// Pure HBM-streaming kernel (512 MiB traffic, ~23 us @ 23.3 TB/s).
// CDNA5 path: async global->LDS B128 copies (ASYNCcnt) double-buffered per
// block, s_wait_asynccnt split counters, non-temporal B128 output stores.
// ---------------------------------------------------------------------------

typedef float v4f __attribute__((ext_vector_type(4)));

#define T_STEPS 4
constexpr int DIMS       = 4096;
constexpr int ROWS       = 16384;
constexpr long long CHUNK_F = (long long)DIMS * (ROWS / T_STEPS); // 16,777,216 floats / chunk
constexpr int CHUNK4     = (int)(CHUNK_F / 4);                    // 4,194,304 vec4 / chunk
constexpr int BLOCK      = 256;
constexpr int KMAX       = CHUNK4 / BLOCK;                        // 16384 block-tiles
constexpr int GRID       = 2048;                                  // persistent; 8 tiles/block

// 16-byte async copy global -> LDS (gfx1250 GLOBAL_LOAD_ASYNC_TO_LDS_B128).
// vdst VGPR carries the wave-relative LDS byte address; HW adds LDS_BASE.
__device__ __forceinline__ void async_copy16(uint64_t gaddr, uint32_t lds_off) {
    asm volatile("global_load_async_to_lds_b128 %0, %1, off"
                 :: "v"(lds_off), "v"(gaddr)
                 : "memory");
}

__device__ __forceinline__ void wait_async_le4() {
    asm volatile("s_wait_asynccnt 0x4" ::: "memory");
}
__device__ __forceinline__ void wait_async_le0() {
    asm volatile("s_wait_asynccnt 0x0" ::: "memory");
}

__global__ __launch_bounds__(BLOCK)
void spike_scan_kernel(const float* __restrict__ in, float* __restrict__ out) {
    // 2 stages x 4 time-slices x 256 lanes x 16B = 32 KB LDS per block.
    __shared__ v4f smem[2][T_STEPS][BLOCK];

    const int      tid   = threadIdx.x;
    const uint32_t sbase = (uint32_t)(uintptr_t)(&smem[0][0][0]); // low 32 bits of
                                                                  // flat LDS addr ==
                                                                  // wave-relative offset
    const uint64_t inb   = (uint64_t)(uintptr_t)in;
    const int      kstride = (int)gridDim.x;

    // Stage all 4 time slices of block-tile k into LDS buffer `buf`.
    auto stage = [&](int buf, int k) {
        const uint64_t col = ((uint64_t)k * BLOCK + (uint64_t)tid) * 16ull; // bytes in chunk
#pragma unroll
        for (int t = 0; t < T_STEPS; ++t) {
            const uint64_t g = inb + (uint64_t)t * (CHUNK_F * 4ull) + col;
            const uint32_t l = sbase +
                (uint32_t)(((buf * T_STEPS + t) * BLOCK + tid) * sizeof(v4f));
            async_copy16(g, l);
        }
    };

    int buf = 0;
    int k   = (int)blockIdx.x;
    if (k < KMAX) stage(buf, k);                 // prime the pipeline

    for (; k < KMAX; k += kstride) {
        const int  kn  = k + kstride;
        const bool pre = (kn < KMAX);            // uniform across the block
        if (pre) stage(buf ^ 1, kn);             // 4 more async loads in flight

        // Loads complete in issue order: <=4 outstanding means the current
        // stage's 4 copies have landed in LDS (only the prefetched stage remains).
        if (pre) wait_async_le4();
        else     wait_async_le0();

        const long long colv = (long long)k * BLOCK + tid;  // vec4 index in chunk
        v4f V = {0.0f, 0.0f, 0.0f, 0.0f};
#pragma unroll
        for (int t = 0; t < T_STEPS; ++t) {
            const v4f I = smem[buf][t][tid];     // ds_load_b128, own slot: no barrier
            V += I;
#pragma unroll
            for (int c = 0; c < 4; ++c)
                V[c] = (__builtin_fabsf(V[c]) > 1.0f) ? 0.0f : V[c];
            // Output is write-once streaming data: non-temporal B128 store so the
            // 192MB L2 keeps serving the *input* stream instead of dead lines.
            v4f* dst = (v4f*)out + (long long)t * CHUNK4 + colv;
            __builtin_nontemporal_store(V, dst);
        }
        buf ^= 1;
    }
}

extern "C" void kernel_launch(void* const* d_in, const int* in_sizes, int n_in,
                              void* d_out, int out_size, void* d_ws, size_t ws_size,
                              hipStream_t stream) {
    const float* in  = (const float*)d_in[0];
    float*       out = (float*)d_out;
    spike_scan_kernel<<<GRID, BLOCK, 0, stream>>>(in, out);
}